// Warp1DOp_4947802325355
// MI455X (gfx1250) — compile-verified
//
#include <hip/hip_runtime.h>
#include <hip/hip_bf16.h>

// Warp1D bilinear-zeros: out[n,c,h,w] = lerp of img[n,c,h,floor(x)], img[..,ceil(x)]
// with x = clip(w - disp[n,0,h,w], -1, W), zero padding outside [0, W).
// Memory-bound gather (~512 MB traffic, ~0.4 GFLOP): disparity math hoisted out
// of the 32-channel loop, 128-bit NT stores, clamped unconditional loads with
// validity folded into the lerp weights. 3D grid removes all div/mod decode.

#define NN 4
#define CC 32
#define HH 384
#define WW 1280
#define W4 (WW / 4)          // 320
#define CH_STRIDE (HH * WW)  // 491,520 elements per channel plane

// Native clang vector type: required by __builtin_nontemporal_store and
// guarantees b128 vector memory ops.
typedef __attribute__((ext_vector_type(4))) float v4f;

__global__ __launch_bounds__(64) void warp1d_bilinear_zeros_kernel(
    const float* __restrict__ img,
    const float* __restrict__ disp,
    float* __restrict__ out)
{
    // grid = (5, HH, NN), block = 64  ->  5*64 == W4 exactly, no tail.
    const int w4 = blockIdx.x * 64 + threadIdx.x;   // [0, 320)
    const int h  = blockIdx.y;                      // [0, 384)
    const int n  = blockIdx.z;                      // [0, 4)
    const int wbase = w4 * 4;

    // One 128-bit disparity load per 4 output pixels (shared by 32 channels).
    const v4f d4 = *(const v4f*)(disp + (n * HH + h) * WW + wbase);

    int   idx0[4], idx1[4];
    float w0[4], w1[4];
#pragma unroll
    for (int j = 0; j < 4; ++j) {
        // Reproduce reference arithmetic exactly: clip to [-1, W], shift +1,
        // floor in padded coordinates, then map back to unpadded indices.
        float xc = (float)(wbase + j) - d4[j];
        xc = fminf(fmaxf(xc, -1.0f), (float)WW);
        const float xp = xc + 1.0f;                 // padded coord in [0, W+1]
        const float xf = floorf(xp);
        const float dx = xp - xf;                   // in [0, 1)
        const int x0 = (int)xf - 1;                 // unpadded floor index
        const int x1 = x0 + 1;                      // == ceil when dx>0; dx==0 gives weight 0
        const float v0 = (x0 >= 0 && x0 < WW) ? 1.0f : 0.0f;
        const float v1 = (x1 >= 0 && x1 < WW) ? 1.0f : 0.0f;
        w0[j] = (1.0f - dx) * v0;                   // zeros padding folded into weights
        w1[j] = dx * v1;
        idx0[j] = min(max(x0, 0), WW - 1);          // clamp -> loads always in-bounds
        idx1[j] = min(max(x1, 0), WW - 1);
    }

    const float* __restrict__ src = img + (size_t)(n * CC * HH + h) * WW;
    float* __restrict__ dst = out + (size_t)(n * CC * HH + h) * WW + wbase;

#pragma unroll 2
    for (int c = 0; c < CC; ++c) {
        // Prefetch next channel's gather region (1.9 MB stride -> no reuse).
        // Speculative: OOB on the last channel is silently dropped.
        __builtin_prefetch(src + CH_STRIDE + idx0[0], 0, 1);

        float a0 = src[idx0[0]], b0 = src[idx1[0]];
        float a1 = src[idx0[1]], b1 = src[idx1[1]];
        float a2 = src[idx0[2]], b2 = src[idx1[2]];
        float a3 = src[idx0[3]], b3 = src[idx1[3]];

        v4f r;
        r[0] = fmaf(w1[0], b0, w0[0] * a0);
        r[1] = fmaf(w1[1], b1, w0[1] * a1);
        r[2] = fmaf(w1[2], b2, w0[2] * a2);
        r[3] = fmaf(w1[3], b3, w0[3] * a3);

        // Write-once output: non-temporal 128-bit store, keep L2 for gathers.
        __builtin_nontemporal_store(r, (v4f*)dst);

        src += CH_STRIDE;
        dst += CH_STRIDE;
    }
}

extern "C" void kernel_launch(void* const* d_in, const int* in_sizes, int n_in,
                              void* d_out, int out_size, void* d_ws, size_t ws_size,
                              hipStream_t stream) {
    const float* img  = (const float*)d_in[0];   // (4, 32, 384, 1280) fp32
    const float* disp = (const float*)d_in[1];   // (4, 1, 384, 1280) fp32
    float* out = (float*)d_out;                  // (4, 32, 384, 1280) fp32

    dim3 grid(W4 / 64, HH, NN);                  // (5, 384, 4) = 7680 blocks
    dim3 block(64, 1, 1);                        // 2 wave32s per block
    warp1d_bilinear_zeros_kernel<<<grid, block, 0, stream>>>(img, disp, out);
}